// CoPLGCF_GCN_36000415875270
// MI455X (gfx1250) — compile-verified
//
#include <hip/hip_runtime.h>
#include <hip/hip_bf16.h>
#include <math.h>

// ---------------------------------------------------------------------------
// CoPLGCF / LightGCN-style forward on MI455X (gfx1250)
//   - SpMM message passing: bandwidth/L2-atomic bound, global_atomic_add_f32
//   - Dense (msg+E) @ W^T + b with V_WMMA_F32_16X16X4_F32 (full f32 precision)
//   - Fused normalize + gather + dot + BCE/reg reduction (wave32 shfl tree)
// ---------------------------------------------------------------------------

typedef __attribute__((ext_vector_type(2))) float v2f;
typedef __attribute__((ext_vector_type(8))) float v8f;

#define D_EMB 64

// Y[rows[e]*64 + j] += scale * vals[e] * X[cols[e]*64 + j]
// 64 consecutive threads handle one edge -> fully coalesced 256B gather and
// 256B of f32 atomics that resolve in the 192MB L2 (msg buffers fit easily).
__global__ void __launch_bounds__(256)
spmm_atomic_kernel(const int* __restrict__ rows, const int* __restrict__ cols,
                   const float* __restrict__ vals, const float* __restrict__ X,
                   float* __restrict__ Y, int nnz, float scale)
{
    long t = (long)blockIdx.x * blockDim.x + threadIdx.x;
    int e = (int)(t >> 6);
    int j = (int)(t & 63);
    if (e >= nnz) return;
    float v = scale * vals[e];
    float x = X[(long)cols[e] * D_EMB + j];
    atomicAdd(&Y[(long)rows[e] * D_EMB + j], v * x);
}

// E_new = leaky_relu((msg + E) @ Wk^T + bk, 0.2), in place on E.
// Block = 256 threads = 8 waves; each wave owns a 16-row tile (M % 16 == 0).
// K=64 consumed in 16 steps of 4 via V_WMMA_F32_16X16X4_F32; W tile in LDS.
__global__ void __launch_bounds__(256)
gemm_lrelu_kernel(float* __restrict__ E, const float* __restrict__ msg,
                  const float* __restrict__ Wk, const float* __restrict__ bk,
                  int M)
{
    __shared__ float Wlds[D_EMB * D_EMB];
    __shared__ float blds[D_EMB];

    int tid = threadIdx.x;
#pragma unroll
    for (int i = 0; i < 16; ++i)
        Wlds[tid + i * 256] = Wk[tid + i * 256];
    if (tid < D_EMB) blds[tid] = bk[tid];
    __syncthreads();

    int wave = tid >> 5;
    int lane = tid & 31;
    long m0 = (long)blockIdx.x * 128 + (long)wave * 16;
    if (m0 >= M) return;             // wave-uniform: EXEC stays all-ones

    int lhalf = lane >> 4;           // 0: K pair {0,1}, 1: K pair {2,3}
    int l16   = lane & 15;
    long arow = m0 + l16;            // A fragment row for this lane
    const float* mrow = msg + arow * D_EMB;
    const float* erow = E   + arow * D_EMB;

    v8f acc[4] = {};                 // 4 N-tiles of 16 -> 16x64 output tile

#pragma unroll
    for (int kk = 0; kk < 16; ++kk) {
        int kbase = kk * 4 + lhalf * 2;
        v2f a;
        a.x = mrow[kbase]     + erow[kbase];
        a.y = mrow[kbase + 1] + erow[kbase + 1];
#pragma unroll
        for (int nt = 0; nt < 4; ++nt) {
            int n = nt * 16 + l16;
            // B(K,N) = W[n][K]  (computing A @ W^T)
            v2f bf;
            bf.x = Wlds[n * D_EMB + kbase];
            bf.y = Wlds[n * D_EMB + kbase + 1];
            acc[nt] = __builtin_amdgcn_wmma_f32_16x16x4_f32(
                false, a, false, bf, (short)0, acc[nt], false, false);
        }
    }

    // C/D layout: VGPR i -> lanes 0-15 row m0+i, lanes 16-31 row m0+i+8
#pragma unroll
    for (int nt = 0; nt < 4; ++nt) {
        int col = nt * 16 + l16;
        float bias = blds[col];
#pragma unroll
        for (int i = 0; i < 8; ++i) {
            long row = m0 + i + lhalf * 8;
            float v = acc[nt][i] + bias;
            v = (v >= 0.0f) ? v : 0.2f * v;
            E[row * D_EMB + col] = v;
        }
    }
}

// One wave per (uid, iid) pair (grid-stride). Lane j covers components j and
// j+32. Normalization of u fused here; wave32 shfl_xor tree reductions.
__global__ void __launch_bounds__(256)
loss_kernel(const float* __restrict__ Eu, const float* __restrict__ Ei,
            const int* __restrict__ uids, const int* __restrict__ iids,
            const float* __restrict__ labels, float* __restrict__ logits,
            float* __restrict__ acc, int B)
{
    int wave   = (int)((blockIdx.x * blockDim.x + threadIdx.x) >> 5);
    int lane   = threadIdx.x & 31;
    int nwaves = (int)((gridDim.x * blockDim.x) >> 5);

    float bce_part = 0.0f, reg_part = 0.0f;

    for (int p = wave; p < B; p += nwaves) {
        long ub = (long)uids[p] * D_EMB;
        long ib = (long)iids[p] * D_EMB;
        float u0 = Eu[ub + lane], u1 = Eu[ub + 32 + lane];
        float w0 = Ei[ib + lane], w1 = Ei[ib + 32 + lane];
        float dot = u0 * w0 + u1 * w1;
        float uu  = u0 * u0 + u1 * u1;
        float vv  = w0 * w0 + w1 * w1;
#pragma unroll
        for (int off = 16; off > 0; off >>= 1) {
            dot += __shfl_xor(dot, off, 32);
            uu  += __shfl_xor(uu,  off, 32);
            vv  += __shfl_xor(vv,  off, 32);
        }
        float un = fmaxf(sqrtf(uu), 1e-12f);
        float logit = dot / un;
        if (lane == 0) {
            logits[p] = logit;
            float lbl = labels[p];
            bce_part += fmaxf(logit, 0.0f) - logit * lbl
                      + log1pf(expf(-fabsf(logit)));
            reg_part += uu / (un * un) + vv;   // ||u_norm||^2 + ||i||^2
        }
    }
    if (lane == 0) {
        atomicAdd(&acc[0], bce_part);
        atomicAdd(&acc[1], reg_part);
    }
}

__global__ void finalize_kernel(const float* __restrict__ acc,
                                float* __restrict__ out, float invB)
{
    out[0] = acc[0] * invB + 1e-6f * acc[1];
}

extern "C" void kernel_launch(void* const* d_in, const int* in_sizes, int n_in,
                              void* d_out, int out_size, void* d_ws, size_t ws_size,
                              hipStream_t stream)
{
    const float* Eu0    = (const float*)d_in[0];
    const float* Ei0    = (const float*)d_in[1];
    const float* W      = (const float*)d_in[2];
    const float* b      = (const float*)d_in[3];
    const int*   pr     = (const int*)d_in[4];
    const int*   pc     = (const int*)d_in[5];
    const float* pv     = (const float*)d_in[6];
    const int*   nr     = (const int*)d_in[7];
    const int*   nc     = (const int*)d_in[8];
    const float* nv     = (const float*)d_in[9];
    const int*   ir     = (const int*)d_in[10];
    const int*   ic     = (const int*)d_in[11];
    const float* iv     = (const float*)d_in[12];
    const int*   uids   = (const int*)d_in[13];
    const int*   iids   = (const int*)d_in[14];
    const float* labels = (const float*)d_in[15];

    const int NU = in_sizes[0] / D_EMB;      // 100000
    const int NI = in_sizes[1] / D_EMB;      // 50000
    const int nnz_pos = in_sizes[4];
    const int nnz_neg = in_sizes[7];
    const int nnz_ii  = in_sizes[10];
    const int Bsz     = in_sizes[13];

    // Workspace layout (floats)
    float* ws    = (float*)d_ws;
    float* Eu    = ws;                                   // NU*64
    float* Ei    = Eu   + (size_t)NU * D_EMB;            // NI*64
    float* msg_u = Ei   + (size_t)NI * D_EMB;            // NU*64
    float* msg_i = msg_u + (size_t)NU * D_EMB;           // NI*64 (contig w/ msg_u)
    float* acc   = msg_i + (size_t)NI * D_EMB;           // 2 floats
    float* logits = (float*)d_out + 1;                   // d_out = [loss, logits]

    hipMemcpyAsync(Eu, Eu0, (size_t)NU * D_EMB * sizeof(float),
                   hipMemcpyDeviceToDevice, stream);
    hipMemcpyAsync(Ei, Ei0, (size_t)NI * D_EMB * sizeof(float),
                   hipMemcpyDeviceToDevice, stream);

    auto spmm = [&](const int* r, const int* c, const float* v,
                    const float* X, float* Y, int nnz, float s) {
        unsigned long long th = (unsigned long long)nnz * D_EMB;
        unsigned blocks = (unsigned)((th + 255) / 256);
        spmm_atomic_kernel<<<blocks, 256, 0, stream>>>(r, c, v, X, Y, nnz, s);
    };

    for (int k = 0; k < 3; ++k) {
        // zero both message buffers (contiguous)
        hipMemsetAsync(msg_u, 0, (size_t)(NU + NI) * D_EMB * sizeof(float), stream);

        spmm(pr, pc, pv, Ei, msg_u, nnz_pos,  1.0f);   // msg_u += pos @ E_i
        spmm(nr, nc, nv, Ei, msg_u, nnz_neg, -1.0f);   // msg_u -= neg @ E_i
        spmm(pc, pr, pv, Eu, msg_i, nnz_pos,  1.0f);   // msg_i += pos^T @ E_u
        spmm(nc, nr, nv, Eu, msg_i, nnz_neg, -1.0f);   // msg_i -= neg^T @ E_u
        spmm(ir, ic, iv, Ei, msg_i, nnz_ii,   1.0f);   // msg_i += II_WEIGHT * ii @ E_i

        gemm_lrelu_kernel<<<(NU + 127) / 128, 256, 0, stream>>>(
            Eu, msg_u, W + (size_t)k * D_EMB * D_EMB, b + (size_t)k * D_EMB, NU);
        gemm_lrelu_kernel<<<(NI + 127) / 128, 256, 0, stream>>>(
            Ei, msg_i, W + (size_t)k * D_EMB * D_EMB, b + (size_t)k * D_EMB, NI);
    }

    hipMemsetAsync(acc, 0, 2 * sizeof(float), stream);
    loss_kernel<<<2048, 256, 0, stream>>>(Eu, Ei, uids, iids, labels,
                                          logits, acc, Bsz);
    finalize_kernel<<<1, 1, 0, stream>>>(acc, (float*)d_out, 1.0f / (float)Bsz);
}